// LinAttention_16226386444902
// MI455X (gfx1250) — compile-verified
//
#include <hip/hip_runtime.h>
#include <hip/hip_bf16.h>

// ---------------------------------------------------------------------------
// LinAttention on MI455X (gfx1250, wave32, WMMA).
//
// Algebraic refactor:  out_b = x_b * M_b  with  M_b = (Wq^T Wk) (x_b^T x_b) Wv^T
// 35 TFLOP naive -> ~2.2 GFLOP + ~51 MB traffic  => memory-bound at 23.3 TB/s.
// Full f32 path via V_WMMA_F32_16X16X4_F32 (no precision loss vs reference).
// Global->LDS staging uses the CDNA5 async path (ASYNCcnt) with double
// buffering so slab copies overlap WMMA compute.
// ---------------------------------------------------------------------------

constexpr int BB    = 32;    // batch
constexpr int NN    = 2048;  // seq len
constexpr int DIMV  = 65;    // feature dim (in == out)
constexpr int DPAD  = 80;    // 65 padded to 5 x 16 WMMA tiles
constexpr int HH    = 64;    // qk hidden
constexpr int NSPL  = 8;     // N-splits for the Gram reduction
constexpr int CHUNK = NN / NSPL;   // 256 rows per workgroup
constexpr int SUBR  = 64;          // rows staged in LDS per iteration
constexpr int NSUB  = CHUNK / SUBR;
constexpr int GSZ   = DPAD * DPAD; // 6400 floats

typedef __attribute__((ext_vector_type(2))) float v2f;
typedef __attribute__((ext_vector_type(8))) float v8f;

typedef __attribute__((address_space(1))) int* gptr_i32;
typedef __attribute__((address_space(3))) int* lptr_i32;

// ---- CDNA5 async global->LDS copy helpers (GLOBAL_LOAD_ASYNC_TO_LDS_B32) ---
__device__ __forceinline__ void async_copy_f32(const float* g, float* l) {
  __builtin_amdgcn_global_load_async_to_lds_b32(
      (gptr_i32)(unsigned long long)(const void*)g,
      (lptr_i32)l, 0, 0);
}
__device__ __forceinline__ void wait_async0() {
  asm volatile("s_wait_asynccnt 0x0" ::: "memory");
}

// ---------------------------------------------------------------------------
// P[d,c] = sum_h Wq[h,d] * Wk[h,c], zero-padded to 80x80. Tiny (one WG).
// ---------------------------------------------------------------------------
__global__ void k_pmat(const float* __restrict__ Wq, const float* __restrict__ Wk,
                       float* __restrict__ P) {
  for (int idx = threadIdx.x; idx < GSZ; idx += blockDim.x) {
    const int d = idx / DPAD, c = idx % DPAD;
    float s = 0.f;
    if (d < DIMV && c < DIMV) {
      for (int h = 0; h < HH; ++h)
        s = fmaf(Wq[h * DIMV + d], Wk[h * DIMV + c], s);
    }
    P[idx] = s;
  }
}

// ---------------------------------------------------------------------------
// Partial Gram: gp[b][split] = x_chunk^T x_chunk  (80x80, padded cols are 0).
// 8 waves/WG; x slabs double-buffered in LDS via async copies; 25 G-tiles;
// WMMA f32 16x16x4 with fully unrolled K chains. For a Gram matrix the A
// (rows d1) and B (cols d2) fragments share one lane layout:
//   lane = {khalf(1b), col(4b)}, v0/v1 = k, k+1.
// ---------------------------------------------------------------------------
__global__ void __launch_bounds__(256)
k_gram(const float* __restrict__ x, float* __restrict__ gp) {
  __shared__ float xs[2][SUBR * DPAD];                 // 2 x 64 x 80 f32 = 40 KB
  const int b     = blockIdx.x;
  const int split = blockIdx.y;
  const int tid   = threadIdx.x;
  const int lane  = tid & 31;
  const int wave  = tid >> 5;                          // 0..7
  const int lhalf = lane >> 4;                         // k-half select
  const int lcol  = lane & 15;                         // row/col within tile

  const v8f zero = {0.f, 0.f, 0.f, 0.f, 0.f, 0.f, 0.f, 0.f};
  v8f acc[4] = {zero, zero, zero, zero};               // up to 4 tiles / wave

  const size_t xbase = ((size_t)b * NN + (size_t)split * CHUNK) * DIMV;

  // Pad columns (c >= 65) are loop-invariant: zero them once in both buffers.
  for (int idx = tid; idx < SUBR * DPAD; idx += 256) {
    if (idx % DPAD >= DIMV) { xs[0][idx] = 0.f; xs[1][idx] = 0.f; }
  }

  // Preload slab 0 (async, ASYNCcnt).
  {
    const float* src = x + xbase;
    for (int idx = tid; idx < SUBR * DIMV; idx += 256) {
      const int r = idx / DIMV, c = idx - r * DIMV;
      async_copy_f32(src + idx, &xs[0][r * DPAD + c]);
    }
  }

  for (int it = 0; it < NSUB; ++it) {
    wait_async0();                 // this wave's copies for slab `it` landed
    __syncthreads();               // all waves' copies (+ pad zeros) visible

    // Kick off slab it+1 while we compute on slab it (double buffer).
    if (it + 1 < NSUB) {
      const float* src = x + xbase + (size_t)(it + 1) * SUBR * DIMV;
      float* dst = xs[(it + 1) & 1];
      for (int idx = tid; idx < SUBR * DIMV; idx += 256) {
        const int r = idx / DIMV, c = idx - r * DIMV;
        async_copy_f32(src + idx, &dst[r * DPAD + c]);
      }
    }

    const float* buf = xs[it & 1];
#pragma unroll
    for (int tt = 0; tt < 4; ++tt) {                   // static acc indexing
      const int t = wave + tt * 8;                     // tile id 0..24
      if (t < 25) {                                    // wave-uniform branch
        const int d1b = (t / 5) * 16;
        const int d2b = (t % 5) * 16;
        v8f c = acc[tt];
#pragma unroll
        for (int k0 = 0; k0 < SUBR; k0 += 4) {
          const int r0 = (k0 + 2 * lhalf) * DPAD;
          v2f a, bv;
          a.x  = buf[r0 + d1b + lcol];
          a.y  = buf[r0 + DPAD + d1b + lcol];
          bv.x = buf[r0 + d2b + lcol];
          bv.y = buf[r0 + DPAD + d2b + lcol];
          c = __builtin_amdgcn_wmma_f32_16x16x4_f32(
              false, a, false, bv, (short)0, c, false, false);
        }
        acc[tt] = c;
      }
    }
    __syncthreads();   // everyone done reading slab `it` before it is reused
  }

  // Store partial G tiles. C/D layout: VGPR j -> row j + 8*lhalf, col = lcol.
  float* g = gp + ((size_t)b * NSPL + split) * GSZ;
#pragma unroll
  for (int tt = 0; tt < 4; ++tt) {
    const int t = wave + tt * 8;
    if (t < 25) {
      const int d1b = (t / 5) * 16;
      const int d2b = (t % 5) * 16;
      union { v8f v; float f[8]; } u;
      u.v = acc[tt];
#pragma unroll
      for (int j = 0; j < 8; ++j)
        g[(d1b + j + 8 * lhalf) * DPAD + d2b + lcol] = u.f[j];
    }
  }
}

// ---------------------------------------------------------------------------
// Per batch: reduce Gram partials (deterministic order), then
//   T = P * G   (80x80),   M[d,e] = sum_k T[d,k] * Wv[e,k]   (Wv^T on right).
// ~1 MFLOP/batch -> plain VALU, one WG per batch.
// ---------------------------------------------------------------------------
__global__ void __launch_bounds__(256)
k_mmat(const float* __restrict__ gp, const float* __restrict__ P,
       const float* __restrict__ Wv, float* __restrict__ M) {
  __shared__ float Gr[GSZ];
  __shared__ float Tl[GSZ];
  const int b   = blockIdx.x;
  const int tid = threadIdx.x;
  const float* g = gp + (size_t)b * NSPL * GSZ;

  for (int idx = tid; idx < GSZ; idx += 256) {
    float s = 0.f;
    for (int p = 0; p < NSPL; ++p) s += g[(size_t)p * GSZ + idx];
    Gr[idx] = s;
  }
  __syncthreads();

  for (int idx = tid; idx < GSZ; idx += 256) {
    const int d = idx / DPAD, e = idx - d * DPAD;
    float s = 0.f;
    for (int k = 0; k < DPAD; ++k) s = fmaf(P[d * DPAD + k], Gr[k * DPAD + e], s);
    Tl[idx] = s;
  }
  __syncthreads();

  for (int idx = tid; idx < GSZ; idx += 256) {
    const int d = idx / DPAD, e = idx - d * DPAD;
    float s = 0.f;
    if (e < DIMV) {
      for (int k = 0; k < DIMV; ++k) s = fmaf(Tl[d * DPAD + k], Wv[e * DIMV + k], s);
    }
    M[(size_t)b * GSZ + idx] = s;                      // padded region = 0
  }
}

// ---------------------------------------------------------------------------
// out_b = x_b * M_b.  Grid (B, N/128); 8 waves/WG, each wave owns one 16-row
// n-tile and all five 16-col e-tiles (A fragment reused 5x). M_b staged in
// LDS via async copies.
// ---------------------------------------------------------------------------
__global__ void __launch_bounds__(256)
k_out(const float* __restrict__ x, const float* __restrict__ M,
      float* __restrict__ out) {
  __shared__ float Ml[GSZ];                            // 25.6 KB
  const int b     = blockIdx.x;
  const int tid   = threadIdx.x;
  const int lane  = tid & 31;
  const int wave  = tid >> 5;
  const int lhalf = lane >> 4;
  const int lcol  = lane & 15;

  const float* Mb = M + (size_t)b * GSZ;
  for (int idx = tid; idx < GSZ; idx += 256)
    async_copy_f32(Mb + idx, &Ml[idx]);
  wait_async0();
  __syncthreads();

  const int nb = (blockIdx.y * 8 + wave) * 16;         // n-tile base row
  const float* xrow = x + ((size_t)b * NN + nb + lcol) * DIMV;

  const v8f zero = {0.f, 0.f, 0.f, 0.f, 0.f, 0.f, 0.f, 0.f};
  v8f acc[5] = {zero, zero, zero, zero, zero};

#pragma unroll
  for (int k0 = 0; k0 < DPAD; k0 += 4) {
    const int c0 = k0 + 2 * lhalf;
    v2f a;
    a.x = (c0 < DIMV) ? xrow[c0] : 0.f;                // zero-pad k >= 65
    a.y = (c0 + 1 < DIMV) ? xrow[c0 + 1] : 0.f;
    const int r0 = c0 * DPAD;
#pragma unroll
    for (int e = 0; e < 5; ++e) {
      v2f bv;
      bv.x = Ml[r0 + e * 16 + lcol];
      bv.y = Ml[r0 + DPAD + e * 16 + lcol];
      acc[e] = __builtin_amdgcn_wmma_f32_16x16x4_f32(
          false, a, false, bv, (short)0, acc[e], false, false);
    }
  }

#pragma unroll
  for (int e = 0; e < 5; ++e) {
    const int col = e * 16 + lcol;
    if (col < DIMV) {
      union { v8f v; float f[8]; } u;
      u.v = acc[e];
#pragma unroll
      for (int j = 0; j < 8; ++j)
        out[((size_t)b * NN + nb + j + 8 * lhalf) * DIMV + col] = u.f[j];
    }
  }
}

// ---------------------------------------------------------------------------
// Launch. Workspace (floats): [Gram partials B*NSPL*GSZ | M B*GSZ | P GSZ]
//   = 1,849,600 f32 = 7.4 MB. All workspace regions are fully written before
//   being read; no float atomics -> bitwise-deterministic across replays.
// ---------------------------------------------------------------------------
extern "C" void kernel_launch(void* const* d_in, const int* in_sizes, int n_in,
                              void* d_out, int out_size, void* d_ws, size_t ws_size,
                              hipStream_t stream) {
  const float* x  = (const float*)d_in[0];
  const float* Wq = (const float*)d_in[1];
  const float* Wk = (const float*)d_in[2];
  const float* Wv = (const float*)d_in[3];
  float* out = (float*)d_out;

  float* ws = (float*)d_ws;
  float* gp = ws;                                   // B * NSPL * GSZ
  float* M  = gp + (size_t)BB * NSPL * GSZ;         // B * GSZ
  float* P  = M + (size_t)BB * GSZ;                 // GSZ

  k_pmat<<<1, 256, 0, stream>>>(Wq, Wk, P);
  k_gram<<<dim3(BB, NSPL), 256, 0, stream>>>(x, gp);
  k_mmat<<<BB, 256, 0, stream>>>(gp, P, Wv, M);
  k_out<<<dim3(BB, NN / 128), 256, 0, stream>>>(x, M, out);
}